// LSTMUsingLSTM_35811437314458
// MI455X (gfx1250) — compile-verified
//
#include <hip/hip_runtime.h>

// ---------------------------------------------------------------------------
// Fused 2-layer LSTM (Keras gate order i,f,g,o) + dense sigmoid head.
// fp32 WMMA (v_wmma_f32_16x16x4_f32), wave32, gfx1250.
//
// - Weight pointers are global-address-space qualified and re-opaqued every
//   timestep (empty asm) so LICM cannot hoist the t-invariant weight loads
//   (no scratch spills) while still emitting global_load (not flat_load,
//   which would also tick DScnt and contend with the LDS pipe).
// ---------------------------------------------------------------------------

#define BATCH 4096
#define SEQ   80
#define EMB   100
#define HID   128
#define G4    512   // 4 * HID
#define BT    32    // batch rows per block
#define XS    108   // LDS stride for x tile   (conflict-free 16-row column reads)
#define HS    132   // LDS stride for h/c tile

typedef __attribute__((ext_vector_type(2))) float v2f;
typedef __attribute__((ext_vector_type(8))) float v8f;
typedef const __attribute__((address_space(1))) float  gfloat;
typedef const __attribute__((address_space(1))) int    gint;

__device__ __forceinline__ float fast_sig(float x) {
  return __builtin_amdgcn_rcpf(1.0f + __expf(-x));
}
__device__ __forceinline__ float fast_tanh(float x) {
  float e = __expf(2.0f * x);
  return (e - 1.0f) * __builtin_amdgcn_rcpf(e + 1.0f);
}

// A-matrix 16x4 f32 fragment (ISA 7.12.2): lane l -> row m=l%16,
// VGPR0 = K + (l<16 ? 0 : 2), VGPR1 = that + 1.
__device__ __forceinline__ v2f load_a_frag(const float* __restrict__ X, int stride,
                                           int mt, int k, int lane) {
  int m  = (lane & 15) + (mt << 4);
  int kk = k + ((lane >> 4) << 1);
  const float* p = X + m * stride + kk;
  v2f a; a.x = p[0]; a.y = p[1];
  return a;
}

// B-matrix 4x16 f32 fragment: lane l -> col n=ncol+l%16,
// VGPR0 = row K + (l<16 ? 0 : 2), VGPR1 = next row.
__device__ __forceinline__ v2f load_b_frag(gfloat* __restrict__ W,
                                           int k, int ncol, int lane) {
  int col = ncol + (lane & 15);
  int kr  = k + ((lane >> 4) << 1);
  v2f b; b.x = W[kr * G4 + col]; b.y = W[(kr + 1) * G4 + col];
  return b;
}

// One LSTM cell step for a 32-row batch tile:
//   z = Xl @ Wg + Hl @ Ug + bg ;  c = sig(f)*c + sig(i)*tanh(g) ; h = sig(o)*tanh(c)
// Wave w owns mt = w>>2 (16-row half) and column blocks nb0, nb0+1 of the
// 128-wide hidden space; for each it accumulates all four gate tiles, so the
// cell update is purely lane-local.
template <int KX>
__device__ __forceinline__ void cell_step(const float* __restrict__ Xl, int xs,
                                          float* __restrict__ Hl,
                                          float* __restrict__ Cl,
                                          gfloat* __restrict__ Wg,
                                          gfloat* __restrict__ Ug,
                                          gfloat* __restrict__ bg,
                                          int wave, int lane) {
  const int mt  = wave >> 2;
  const int nb0 = (wave << 1) & 7;

  v8f acc[2][4];
  const v8f vz = {0.f, 0.f, 0.f, 0.f, 0.f, 0.f, 0.f, 0.f};
#pragma unroll
  for (int c = 0; c < 2; ++c)
#pragma unroll
    for (int g = 0; g < 4; ++g) acc[c][g] = vz;

  // ---- X @ W ----
  for (int k = 0; k < KX; k += 4) {
    v2f a = load_a_frag(Xl, xs, mt, k, lane);
    v2f bfr[2][4];
#pragma unroll
    for (int c = 0; c < 2; ++c)
#pragma unroll
      for (int g = 0; g < 4; ++g)
        bfr[c][g] = load_b_frag(Wg, k, (nb0 + c) * 16 + g * HID, lane);
#pragma unroll
    for (int c = 0; c < 2; ++c)
#pragma unroll
      for (int g = 0; g < 4; ++g)
        acc[c][g] = __builtin_amdgcn_wmma_f32_16x16x4_f32(
            false, a, false, bfr[c][g], (short)0, acc[c][g], false, false);
  }
  // ---- H @ U ----
  for (int k = 0; k < HID; k += 4) {
    v2f a = load_a_frag(Hl, HS, mt, k, lane);
    v2f bfr[2][4];
#pragma unroll
    for (int c = 0; c < 2; ++c)
#pragma unroll
      for (int g = 0; g < 4; ++g)
        bfr[c][g] = load_b_frag(Ug, k, (nb0 + c) * 16 + g * HID, lane);
#pragma unroll
    for (int c = 0; c < 2; ++c)
#pragma unroll
      for (int g = 0; g < 4; ++g)
        acc[c][g] = __builtin_amdgcn_wmma_f32_16x16x4_f32(
            false, a, false, bfr[c][g], (short)0, acc[c][g], false, false);
  }

  __syncthreads();  // all waves done reading Hl before anyone rewrites it

  // ---- gates + state update (each (m,n) element owned by exactly one lane) ----
#pragma unroll
  for (int c = 0; c < 2; ++c) {
    int n  = (nb0 + c) * 16 + (lane & 15);
    float bi = bg[n];
    float bf = bg[HID + n];
    float bc = bg[2 * HID + n];
    float bo = bg[3 * HID + n];
#pragma unroll
    for (int j = 0; j < 8; ++j) {
      int m = (mt << 4) + j + ((lane >> 4) << 3);  // C/D layout: VGPR j -> M = j (+8 hi lanes)
      float ig = fast_sig (acc[c][0][j] + bi);
      float fg = fast_sig (acc[c][1][j] + bf);
      float gg = fast_tanh(acc[c][2][j] + bc);
      float og = fast_sig (acc[c][3][j] + bo);
      float cold = Cl[m * HS + n];
      float cnew = fg * cold + ig * gg;
      Cl[m * HS + n] = cnew;
      Hl[m * HS + n] = og * fast_tanh(cnew);
    }
  }
  __syncthreads();
}

__global__ __launch_bounds__(256, 1)
void lstm2_fused_kernel(const int*   __restrict__ tokens_p,
                        const float* __restrict__ emb_p,
                        const float* __restrict__ W1g, const float* __restrict__ U1g,
                        const float* __restrict__ b1p,
                        const float* __restrict__ W2g, const float* __restrict__ U2g,
                        const float* __restrict__ b2p,
                        const float* __restrict__ Wd, const float* __restrict__ bd,
                        float* __restrict__ out) {
  __shared__ float Xl[BT * XS];
  __shared__ float H1[BT * HS];
  __shared__ float C1[BT * HS];
  __shared__ float H2[BT * HS];
  __shared__ float C2[BT * HS];

  const int tid  = threadIdx.x;
  const int lane = tid & 31;
  const int wave = tid >> 5;
  const int row0 = blockIdx.x * BT;

  for (int i = tid; i < BT * HS; i += 256) {
    H1[i] = 0.f; C1[i] = 0.f; H2[i] = 0.f; C2[i] = 0.f;
  }
  __syncthreads();

  // Global-address-space weight pointers; re-opaqued every timestep.
  gfloat* W1     = (gfloat*)W1g;
  gfloat* U1     = (gfloat*)U1g;
  gfloat* W2     = (gfloat*)W2g;
  gfloat* U2     = (gfloat*)U2g;
  gfloat* b1     = (gfloat*)b1p;
  gfloat* b2     = (gfloat*)b2p;
  gfloat* emb    = (gfloat*)emb_p;
  gint*   tokens = (gint*)tokens_p;

  for (int t = 0; t < SEQ; ++t) {
    // Defeat loop-invariant hoisting of the (t-invariant) weight loads:
    // without this, the compiler preloads ~1 MB of weights and spills to
    // scratch, putting a scratch reload + full wait in front of every WMMA.
    asm volatile("" : "+s"(W1), "+s"(U1), "+s"(W2), "+s"(U2));

    // Gather this timestep's embeddings into LDS (32 rows x 100 f32).
    for (int idx = tid; idx < BT * EMB; idx += 256) {
      int r = idx / EMB;
      int e = idx - r * EMB;
      int tok = tokens[(row0 + r) * SEQ + t];
      Xl[r * XS + e] = emb[tok * EMB + e];
    }
    __syncthreads();

    cell_step<EMB>(Xl, XS, H1, C1, W1, U1, b1, wave, lane);  // layer 1
    cell_step<HID>(H1, HS, H2, C2, W2, U2, b2, wave, lane);  // layer 2 (input = h1)
  }

  // Dense head: out = sigmoid(h2 @ Wd + bd), one row per lane of wave 0.
  if (tid < BT) {
    float s = bd[0];
#pragma unroll 8
    for (int k = 0; k < HID; ++k) s += H2[tid * HS + k] * Wd[k];
    out[row0 + tid] = fast_sig(s);
  }
}

extern "C" void kernel_launch(void* const* d_in, const int* in_sizes, int n_in,
                              void* d_out, int out_size, void* d_ws, size_t ws_size,
                              hipStream_t stream) {
  (void)in_sizes; (void)n_in; (void)out_size; (void)d_ws; (void)ws_size;
  const int*   tokens = (const int*)  d_in[0];
  const float* emb    = (const float*)d_in[1];
  const float* W1     = (const float*)d_in[2];
  const float* U1     = (const float*)d_in[3];
  const float* b1     = (const float*)d_in[4];
  const float* W2     = (const float*)d_in[5];
  const float* U2     = (const float*)d_in[6];
  const float* b2     = (const float*)d_in[7];
  const float* Wd     = (const float*)d_in[8];
  const float* bd     = (const float*)d_in[9];
  float* out = (float*)d_out;

  dim3 grid(BATCH / BT);   // 128 blocks
  dim3 block(256);         // 8 waves (wave32)
  lstm2_fused_kernel<<<grid, block, 0, stream>>>(tokens, emb, W1, U1, b1,
                                                 W2, U2, b2, Wd, bd, out);
}